// HAN_39633958208188
// MI455X (gfx1250) — compile-verified
//
#include <hip/hip_runtime.h>
#include <hip/hip_bf16.h>
#include <math.h>

#define H_   8
#define D_   16
#define HID_ 128
#define N_   200000
#define E_   500000
#define IN_  256
#define CLS_ 3

typedef __attribute__((ext_vector_type(2))) float v2f;
typedef __attribute__((ext_vector_type(8))) float v8f;

// ---------------------------------------------------------------- utilities

__global__ void fill_f32(float* __restrict__ p, float v, int n) {
    int i = blockIdx.x * blockDim.x + threadIdx.x;
    if (i < n) p[i] = v;
}

__global__ void relu_inplace(float* __restrict__ p, int n) {
    int i = blockIdx.x * blockDim.x + threadIdx.x;
    if (i < n) p[i] = fmaxf(p[i], 0.0f);
}

// Native CDNA5 float atomic max: lowers to GLOBAL_ATOMIC_MAX_NUM_F32
// (IEEE maximumNumber; correct for our finite alphas vs -1e30 init).
__device__ __forceinline__ void atomic_max_f32(float* addr, float val) {
    __hip_atomic_fetch_max(addr, val, __ATOMIC_RELAXED, __HIP_MEMORY_SCOPE_AGENT);
}

// ---------------------------------------------------------------- WMMA GEMM
// C[M,128] = A[M,K] @ B[K,128] + bias[128]
// One wave computes one 16x16 output tile with V_WMMA_F32_16X16X4_F32.
// blockDim = (32, 8): the 8 waves of a block cover one 16-row x 128-col strip.
// mode 0: store C + bias
// mode 1: atomicAdd(score, sum(tanh(C+bias) * q[col]))  (tile never stored)
__global__ void gemm_wmma_f32(const float* __restrict__ A,
                              const float* __restrict__ B,
                              const float* __restrict__ bias,
                              float* __restrict__ C,
                              int K, int mode,
                              const float* __restrict__ q,
                              float* __restrict__ score) {
    const int lane  = threadIdx.x;            // 0..31
    const int ntile = threadIdx.y;            // 0..7 column tile
    const int row0  = blockIdx.x * 16;
    const int col0  = ntile * 16;
    const bool hi   = lane >= 16;
    const int  l16  = lane & 15;
    const int  arow = row0 + l16;             // A row this lane feeds
    const int  bcol = col0 + l16;             // B/C column this lane feeds

    v8f acc = {};
    for (int k = 0; k < K; k += 4) {
        // 16x4 f32 A layout: lanes 0-15 hold K = k,k+1; lanes 16-31 hold K = k+2,k+3
        const int kk = k + (hi ? 2 : 0);
        const float2 av = *(const float2*)(A + (size_t)arow * K + kk);
        v2f a; a.x = av.x; a.y = av.y;
        // 4x16 f32 B layout mirrors A: same K split across lane halves
        v2f b;
        b.x = B[(size_t)(kk    ) * HID_ + bcol];
        b.y = B[(size_t)(kk + 1) * HID_ + bcol];
        acc = __builtin_amdgcn_wmma_f32_16x16x4_f32(
                false, a, false, b, (short)0, acc, false, false);
    }

    const float bv = bias[bcol];
    if (mode == 0) {
        // C/D layout: lanes 0-15 -> rows row0+0..7 ; lanes 16-31 -> rows row0+8..15
#pragma unroll
        for (int r = 0; r < 8; ++r) {
            const int row = row0 + r + (hi ? 8 : 0);
            C[(size_t)row * HID_ + bcol] = acc[r] + bv;
        }
    } else {
        const float qv = q[bcol];
        float s = 0.0f;
#pragma unroll
        for (int r = 0; r < 8; ++r) s += tanhf(acc[r] + bv) * qv;
#pragma unroll
        for (int off = 16; off > 0; off >>= 1) s += __shfl_down(s, off, 32);
        if (lane == 0) atomicAdd(score, s);
    }
}

// ---------------------------------------------------------------- attention

// per (node, head): a_s = <h, att_src>, a_d = <h, att_dst> for both relations
__global__ void attn_coef(const float* __restrict__ h,
                          const float* __restrict__ v_s0, const float* __restrict__ v_d0,
                          const float* __restrict__ v_s1, const float* __restrict__ v_d1,
                          float* __restrict__ as0, float* __restrict__ ad0,
                          float* __restrict__ as1, float* __restrict__ ad1) {
    int i = blockIdx.x * blockDim.x + threadIdx.x;   // i = n*H + head
    if (i >= N_ * H_) return;
    const int hh = i & (H_ - 1);
    const float* hp = h + (size_t)i * D_;
    float s0 = 0.f, d0 = 0.f, s1 = 0.f, d1 = 0.f;
#pragma unroll
    for (int j = 0; j < D_; ++j) {
        const float v = hp[j];
        s0 += v * v_s0[hh * D_ + j];
        d0 += v * v_d0[hh * D_ + j];
        s1 += v * v_s1[hh * D_ + j];
        d1 += v * v_d1[hh * D_ + j];
    }
    as0[i] = s0; ad0[i] = d0; as1[i] = s1; ad1[i] = d1;
}

// pass 1: alpha = leaky_relu(a_s[src] + a_d[dst], 0.2); segment max over dst
__global__ void edge_max(const int* __restrict__ src, const int* __restrict__ dst,
                         const float* __restrict__ as, const float* __restrict__ ad,
                         float* __restrict__ alpha, float* __restrict__ m) {
    int i = blockIdx.x * blockDim.x + threadIdx.x;   // i = e*H + head
    if (i >= E_ * H_) return;
    const int e = i >> 3, hh = i & 7;
    const int s = src[e], d = dst[e];
    float al = as[s * H_ + hh] + ad[d * H_ + hh];
    al = al > 0.0f ? al : 0.2f * al;
    alpha[i] = al;
    atomic_max_f32(&m[d * H_ + hh], al);
}

// pass 2: ex = exp(alpha - m[dst]); segment sum over dst (alpha overwritten)
__global__ void edge_expsum(const int* __restrict__ dst,
                            const float* __restrict__ m,
                            float* __restrict__ alpha, float* __restrict__ den) {
    int i = blockIdx.x * blockDim.x + threadIdx.x;
    if (i >= E_ * H_) return;
    const int e = i >> 3, hh = i & 7;
    const int d = dst[e];
    const float ex = expf(alpha[i] - m[d * H_ + hh]);
    alpha[i] = ex;
    atomicAdd(&den[d * H_ + hh], ex);
}

// pass 3: out[dst] += h[src] * (ex / (den[dst] + eps))
__global__ void edge_scatter(const int* __restrict__ src, const int* __restrict__ dst,
                             const float* __restrict__ ex, const float* __restrict__ den,
                             const float* __restrict__ h, float* __restrict__ out) {
    int i = blockIdx.x * blockDim.x + threadIdx.x;
    if (i >= E_ * H_) return;
    const int e = i >> 3, hh = i & 7;
    const int s = src[e], d = dst[e];
    const float w = ex[i] / (den[d * H_ + hh] + 1e-16f);
    const float* hp = h + (size_t)s * HID_ + hh * D_;
    float* op = out + (size_t)d * HID_ + hh * D_;
#pragma unroll
    for (int j = 0; j < D_; ++j) atomicAdd(&op[j], hp[j] * w);
}

// ---------------------------------------------------------------- classifier

// one block (128 threads) per node: beta-softmax fuse + [128]x[128x3] linear
__global__ void final_kernel(const float* __restrict__ out0,
                             const float* __restrict__ out1,
                             const float* __restrict__ score,
                             const float* __restrict__ lin_w,
                             const float* __restrict__ lin_b,
                             float* __restrict__ y) {
    __shared__ float sdata[CLS_ * HID_];
    const int n = blockIdx.x;
    const int f = threadIdx.x;                        // 0..127
    const float s0 = score[0] * (1.0f / (float)N_);
    const float s1 = score[1] * (1.0f / (float)N_);
    const float mx = fmaxf(s0, s1);
    const float e0 = expf(s0 - mx), e1 = expf(s1 - mx);
    const float b0 = e0 / (e0 + e1), b1 = e1 / (e0 + e1);
    const float fused = b0 * out0[(size_t)n * HID_ + f] + b1 * out1[(size_t)n * HID_ + f];
#pragma unroll
    for (int c = 0; c < CLS_; ++c) sdata[c * HID_ + f] = fused * lin_w[f * CLS_ + c];
    __syncthreads();
    for (int s = HID_ / 2; s > 0; s >>= 1) {
        if (f < s) {
#pragma unroll
            for (int c = 0; c < CLS_; ++c) sdata[c * HID_ + f] += sdata[c * HID_ + f + s];
        }
        __syncthreads();
    }
    if (f < CLS_) y[(size_t)n * CLS_ + f] = sdata[f * HID_] + lin_b[f];
}

// ---------------------------------------------------------------- launcher

extern "C" void kernel_launch(void* const* d_in, const int* in_sizes, int n_in,
                              void* d_out, int out_size, void* d_ws, size_t ws_size,
                              hipStream_t stream) {
    (void)in_sizes; (void)n_in; (void)out_size; (void)ws_size;

    const float* x        = (const float*)d_in[0];
    const float* W_proj   = (const float*)d_in[1];
    const float* b_proj   = (const float*)d_in[2];
    const float* att_src0 = (const float*)d_in[3];
    const float* att_dst0 = (const float*)d_in[4];
    const float* att_src1 = (const float*)d_in[5];
    const float* att_dst1 = (const float*)d_in[6];
    const float* q        = (const float*)d_in[7];
    const float* k_w      = (const float*)d_in[8];
    const float* k_b      = (const float*)d_in[9];
    const float* lin_w    = (const float*)d_in[10];
    const float* lin_b    = (const float*)d_in[11];
    const int*   e0_src   = (const int*)d_in[12];
    const int*   e0_dst   = (const int*)d_in[13];
    const int*   e1_src   = (const int*)d_in[14];
    const int*   e1_dst   = (const int*)d_in[15];
    float* y = (float*)d_out;

    // workspace carve-out (floats)
    float* ws   = (float*)d_ws;
    float* h    = ws; ws += (size_t)N_ * HID_;
    float* out0 = ws; ws += (size_t)N_ * HID_;
    float* out1 = ws; ws += (size_t)N_ * HID_;
    float* as0  = ws; ws += (size_t)N_ * H_;
    float* ad0  = ws; ws += (size_t)N_ * H_;
    float* as1  = ws; ws += (size_t)N_ * H_;
    float* ad1  = ws; ws += (size_t)N_ * H_;
    float* m0   = ws; ws += (size_t)N_ * H_;
    float* den0 = ws; ws += (size_t)N_ * H_;
    float* m1   = ws; ws += (size_t)N_ * H_;
    float* den1 = ws; ws += (size_t)N_ * H_;
    float* alpha = ws; ws += (size_t)E_ * H_;   // reused for both relations
    float* score = ws; ws += 2;

    const int NH  = N_ * HID_;    // 25.6M
    const int NHh = N_ * H_;      // 1.6M
    const int EH  = E_ * H_;      // 4.0M
    const int BS  = 256;

    // init accumulators
    fill_f32<<<(NH  + BS - 1) / BS, BS, 0, stream>>>(out0, 0.0f, NH);
    fill_f32<<<(NH  + BS - 1) / BS, BS, 0, stream>>>(out1, 0.0f, NH);
    fill_f32<<<(NHh + BS - 1) / BS, BS, 0, stream>>>(m0, -1e30f, NHh);
    fill_f32<<<(NHh + BS - 1) / BS, BS, 0, stream>>>(den0, 0.0f, NHh);
    fill_f32<<<(NHh + BS - 1) / BS, BS, 0, stream>>>(m1, -1e30f, NHh);
    fill_f32<<<(NHh + BS - 1) / BS, BS, 0, stream>>>(den1, 0.0f, NHh);
    fill_f32<<<1, 2, 0, stream>>>(score, 0.0f, 2);

    // projection GEMM: h = x @ W_proj + b_proj   (WMMA f32 16x16x4)
    dim3 gblk(32, HID_ / 16);
    gemm_wmma_f32<<<N_ / 16, gblk, 0, stream>>>(x, W_proj, b_proj, h, IN_, 0,
                                                nullptr, nullptr);

    // attention coefficients for both relations
    attn_coef<<<(NHh + BS - 1) / BS, BS, 0, stream>>>(h, att_src0, att_dst0,
                                                      att_src1, att_dst1,
                                                      as0, ad0, as1, ad1);

    // relation 0
    edge_max    <<<(EH + BS - 1) / BS, BS, 0, stream>>>(e0_src, e0_dst, as0, ad0, alpha, m0);
    edge_expsum <<<(EH + BS - 1) / BS, BS, 0, stream>>>(e0_dst, m0, alpha, den0);
    edge_scatter<<<(EH + BS - 1) / BS, BS, 0, stream>>>(e0_src, e0_dst, alpha, den0, h, out0);

    // relation 1 (alpha buffer reused; stream-ordered)
    edge_max    <<<(EH + BS - 1) / BS, BS, 0, stream>>>(e1_src, e1_dst, as1, ad1, alpha, m1);
    edge_expsum <<<(EH + BS - 1) / BS, BS, 0, stream>>>(e1_dst, m1, alpha, den1);
    edge_scatter<<<(EH + BS - 1) / BS, BS, 0, stream>>>(e1_src, e1_dst, alpha, den1, h, out1);

    relu_inplace<<<(NH + BS - 1) / BS, BS, 0, stream>>>(out0, NH);
    relu_inplace<<<(NH + BS - 1) / BS, BS, 0, stream>>>(out1, NH);

    // semantic attention scores: sum_n tanh(out_m @ k_w + k_b) . q  (WMMA + fused reduce)
    gemm_wmma_f32<<<N_ / 16, gblk, 0, stream>>>(out0, k_w, k_b, nullptr, HID_, 1, q, score + 0);
    gemm_wmma_f32<<<N_ / 16, gblk, 0, stream>>>(out1, k_w, k_b, nullptr, HID_, 1, q, score + 1);

    // fuse with softmax(beta) and apply classifier
    final_kernel<<<N_, HID_, 0, stream>>>(out0, out1, score, lin_w, lin_b, y);
}